// PoseNMSAndReturnAsBatchedResult_5153960755741
// MI455X (gfx1250) — compile-verified
//
#include <hip/hip_runtime.h>

// Problem constants (match the reference)
#define NBATCH  64
#define NPRE    8192
#define MMAX    128
#define NJ      17
#define SELPER  64
#define NSEL    (NBATCH * SELPER)        // 4096
#define JF      (NJ * 3)                 // 51 floats per joints row (204 bytes)

// Flat float offsets inside d_out (outputs concatenated in return order)
#define OFF_NUM     0
#define OFF_BOXES   (NBATCH)                          // 64
#define OFF_SCORES  (OFF_BOXES + NBATCH * MMAX * 4)   // 32832
#define OFF_JOINTS  (OFF_SCORES + NBATCH * MMAX)      // 41024
// total = 458816 floats

// ---------------------------------------------------------------------------
// Kernel 1: zero the whole output (vectorized float4 stores, grid-stride safe)
// ---------------------------------------------------------------------------
__global__ void k_zero(float* __restrict__ out, int n) {
    int i = blockIdx.x * blockDim.x + threadIdx.x;
    int n4 = n >> 2;
    if (i < n4) {
        ((float4*)out)[i] = make_float4(0.f, 0.f, 0.f, 0.f);
    }
    if (i < (n & 3)) {
        out[(n4 << 2) + i] = 0.f;
    }
}

// ---------------------------------------------------------------------------
// Kernel 2: stable per-batch rank (pos) + num_predictions.
// One thread per batch scans selections IN ORDER -> exact cumsum semantics,
// fully deterministic (no atomics). 64 threads, 4096 iterations (broadcast
// loads, L2/L0 resident) -- negligible.
// ---------------------------------------------------------------------------
__global__ void k_pos(const int* __restrict__ sel,
                      int* __restrict__ pos,
                      float* __restrict__ out_num) {
    int j = threadIdx.x;                  // batch id 0..63
    int cnt = 0;
    for (int s = 0; s < NSEL; ++s) {
        int bs = sel[3 * s];
        bs = min(max(bs, 0), NBATCH - 1); // gather-clip semantics
        if (bs == j) { pos[s] = cnt++; }
    }
    out_num[j] = (float)cnt;              // reference dtype int32 -> d_out is float
}

// ---------------------------------------------------------------------------
// Kernel 3: gather/scatter boxes (float4) + scores (float).
// One thread per selection. Prefetch the joints row into L2 (192 MB) so the
// joints kernel hits cache: emits global_prefetch_b8 on gfx1250.
// ---------------------------------------------------------------------------
__global__ void k_boxes_scores(const float4* __restrict__ boxes4,
                               const float*  __restrict__ scores,
                               const float*  __restrict__ joints,
                               const int*    __restrict__ sel,
                               const int*    __restrict__ pos,
                               float4* __restrict__ oboxes4,
                               float*  __restrict__ oscores) {
    int s = blockIdx.x * blockDim.x + threadIdx.x;
    if (s >= NSEL) return;
    int b = sel[3 * s + 0];
    int x = sel[3 * s + 2];
    b = min(max(b, 0), NBATCH - 1);
    x = min(max(x, 0), NPRE - 1);
    // l clamps to 0 (pred_scores last dim == 1)

    // Warm L2 for the joints pass (row spans 204 B -> two 128B lines + tail)
    const char* jrow = (const char*)(joints + (size_t)(b * NPRE + x) * JF);
    __builtin_prefetch(jrow, 0, 1);
    __builtin_prefetch(jrow + 128, 0, 1);

    int p = pos[s];
    if (p < 0 || p >= MMAX) return;       // scatter mode='drop'
    oboxes4[b * MMAX + p] = boxes4[b * NPRE + x];
    oscores[b * MMAX + p] = scores[b * NPRE + x];
}

// ---------------------------------------------------------------------------
// Kernel 4: joints rows (51 floats / 204 B each) via the CDNA5 async
// global<->LDS engine. One wave (32 lanes) per selection:
//   global_load_async_to_lds_b32   (memory -> LDS, ASYNCcnt)
//   s_wait_asynccnt 0
//   global_store_async_from_lds_b32 (LDS -> memory, ASYNCcnt)
// Per-wave 256 B LDS staging slot; addresses are wave-uniform so the 64-bit
// bases go through "s" constraints (readfirstlane'd by the compiler).
// ---------------------------------------------------------------------------
__global__ void k_joints(const float* __restrict__ joints,
                         const int*   __restrict__ sel,
                         const int*   __restrict__ pos,
                         float* __restrict__ ojoints) {
    __shared__ float stage[8 * 64];       // 8 waves/block * 64 dwords (pad)
    const int lane = threadIdx.x & 31;
    const int wv   = threadIdx.x >> 5;
    const int s    = blockIdx.x * 8 + wv;
    if (s >= NSEL) return;

    int b = sel[3 * s + 0];
    int x = sel[3 * s + 2];
    b = min(max(b, 0), NBATCH - 1);
    x = min(max(x, 0), NPRE - 1);
    int p = pos[s];
    if (p < 0 || p >= MMAX) return;       // drop (wave-uniform branch)

    unsigned long long src =
        (unsigned long long)(const void*)(joints + (size_t)(b * NPRE + x) * JF);
    unsigned long long dst =
        (unsigned long long)(void*)(ojoints + ((size_t)b * MMAX + p) * JF);
    // Low 32 bits of the generic address == LDS byte offset (ISA 10.2).
    unsigned lds_base = (unsigned)(size_t)(&stage[wv * 64]);

    // memory -> LDS (51 dwords: round 0 lanes 0..31, round 1 lanes 0..18)
    #pragma unroll
    for (int k = 0; k < 2; ++k) {
        int d = k * 32 + lane;
        if (d < JF) {
            unsigned ldsa = lds_base + d * 4;
            unsigned voff = d * 4;
            asm volatile("global_load_async_to_lds_b32 %0, %1, %2"
                         :: "v"(ldsa), "v"(voff), "s"(src) : "memory");
        }
    }
    asm volatile("s_wait_asynccnt 0" ::: "memory");

    // LDS -> memory
    #pragma unroll
    for (int k = 0; k < 2; ++k) {
        int d = k * 32 + lane;
        if (d < JF) {
            unsigned ldsa = lds_base + d * 4;
            unsigned voff = d * 4;
            asm volatile("global_store_async_from_lds_b32 %0, %1, %2"
                         :: "v"(voff), "v"(ldsa), "s"(dst) : "memory");
        }
    }
    asm volatile("s_wait_asynccnt 0" ::: "memory");
}

// ---------------------------------------------------------------------------
// Launcher
// ---------------------------------------------------------------------------
extern "C" void kernel_launch(void* const* d_in, const int* in_sizes, int n_in,
                              void* d_out, int out_size, void* d_ws, size_t ws_size,
                              hipStream_t stream) {
    const float* boxes  = (const float*)d_in[0];  // (64, 8192, 4)
    const float* scores = (const float*)d_in[1];  // (64, 8192, 1)
    const float* joints = (const float*)d_in[2];  // (64, 8192, 17, 3)
    const int*   sel    = (const int*)  d_in[3];  // (4096, 3)
    float* out = (float*)d_out;
    int*   pos = (int*)d_ws;                      // 4096 ints of scratch

    // 1) zero-fill all outputs
    {
        int n4 = (out_size >> 2);
        int blocks = (n4 + 255) / 256;
        if (blocks < 1) blocks = 1;
        k_zero<<<blocks, 256, 0, stream>>>(out, out_size);
    }
    // 2) stable ranks + num_predictions
    k_pos<<<1, NBATCH, 0, stream>>>(sel, pos, out + OFF_NUM);
    // 3) boxes + scores (+ L2 prefetch of joints rows)
    k_boxes_scores<<<(NSEL + 255) / 256, 256, 0, stream>>>(
        (const float4*)boxes, scores, joints, sel, pos,
        (float4*)(out + OFF_BOXES), out + OFF_SCORES);
    // 4) joints rows via async global<->LDS (one wave per selection)
    k_joints<<<NSEL / 8, 256, 0, stream>>>(joints, sel, pos, out + OFF_JOINTS);
}